// ConvNormBlock_29360396435530
// MI455X (gfx1250) — compile-verified
//
#include <hip/hip_runtime.h>
#include <hip/hip_bf16.h>

typedef __attribute__((ext_vector_type(2))) float v2f;
typedef __attribute__((ext_vector_type(8))) float v8f;

#define D_IN   256
#define D_OUT  512
#define BN_EPS 1e-5f

// ---------------------------------------------------------------- degree prep
__global__ __launch_bounds__(256) void deg_init_kernel(float* __restrict__ deg, int N) {
    int i = blockIdx.x * blockDim.x + threadIdx.x;
    if (i < N) deg[i] = 1.0f;  // self-loop weight 1
}

__global__ __launch_bounds__(512) void stats_init_kernel(float* __restrict__ colsum,
                                                         float* __restrict__ colsumsq) {
    int i = threadIdx.x;  // blockDim = 512
    colsum[i]   = 0.0f;
    colsumsq[i] = 0.0f;
}

__global__ __launch_bounds__(256) void deg_scatter_kernel(const int* __restrict__ ei,
                                                          const float* __restrict__ ew,
                                                          float* __restrict__ deg,
                                                          int E) {
    int e = blockIdx.x * blockDim.x + threadIdx.x;
    if (e < E) {
        int c = ei[E + e];                 // target node
        atomicAdd(&deg[c], ew[e]);
    }
}

__global__ __launch_bounds__(256) void deg_invsqrt_kernel(float* __restrict__ deg, int N) {
    int i = blockIdx.x * blockDim.x + threadIdx.x;
    if (i < N) {
        float d = deg[i];
        deg[i] = (d > 0.0f) ? rsqrtf(d) : 0.0f;
    }
}

// ---------------------------------------------------------------- fp32 WMMA GEMM
// xw[M,512] = x[M,256] @ W[256,512].  One wave computes a 64(M) x 64(N) tile:
// 4 A-fragments x 4 B-fragments -> 16 independent V_WMMA_F32_16X16X4_F32
// accumulator chains per K-step (K stepped by 4).  Out-of-range rows are
// handled by clamping the A row pointer (their accumulator slots are never
// stored), keeping the inner loop free of per-iteration selects.
__global__ __launch_bounds__(32) void gemm_wmma_kernel(const float* __restrict__ x,
                                                       const float* __restrict__ W,
                                                       float* __restrict__ xw,
                                                       int M) {
    const int m0   = blockIdx.x * 64;
    const int n0   = blockIdx.y * 64;
    const int lane = threadIdx.x;      // 0..31
    const int half = lane >> 4;        // 0: K pair {k,k+1}, 1: K pair {k+2,k+3}
    const int lm   = lane & 15;

    // Clamped per-tile A row pointers (A row M only feeds D row M).
    const float* arow[4];
    #pragma unroll
    for (int t = 0; t < 4; ++t) {
        int r = m0 + 16 * t + lm;
        if (r >= M) r = M - 1;
        arow[t] = x + (size_t)r * D_IN + 2 * half;
    }

    v8f acc[4][4];
    #pragma unroll
    for (int t = 0; t < 4; ++t)
        #pragma unroll
        for (int j = 0; j < 4; ++j)
            acc[t][j] = (v8f){};

    for (int k = 0; k < D_IN; k += 4) {
        // A fragments: 16x4 fp32; lane holds (row, k+2*half .. k+2*half+1)
        v2f a[4];
        #pragma unroll
        for (int t = 0; t < 4; ++t)
            a[t] = *(const v2f*)(arow[t] + k);

        // B fragments: 4x16 fp32 per n-tile; VGPR0 = K{k+2h}, VGPR1 = K{k+2h+1}
        const float* bp = W + (size_t)(k + 2 * half) * D_OUT + n0 + lm;
        v2f b[4];
        #pragma unroll
        for (int j = 0; j < 4; ++j) {
            b[j].x = bp[16 * j];
            b[j].y = bp[D_OUT + 16 * j];
        }

        #pragma unroll
        for (int t = 0; t < 4; ++t)
            #pragma unroll
            for (int j = 0; j < 4; ++j)
                acc[t][j] = __builtin_amdgcn_wmma_f32_16x16x4_f32(
                    false, a[t], false, b[j], (short)0, acc[t][j], false, false);
    }

    // C/D layout: VGPR v -> M = v + 8*half, N = lm
    #pragma unroll
    for (int t = 0; t < 4; ++t) {
        #pragma unroll
        for (int v = 0; v < 8; ++v) {
            int rr = m0 + 16 * t + v + 8 * half;
            if (rr < M) {
                float* o = xw + (size_t)rr * D_OUT + n0 + lm;
                o[0]  = acc[t][0][v];
                o[16] = acc[t][1][v];
                o[32] = acc[t][2][v];
                o[48] = acc[t][3][v];
            }
        }
    }
}

// ---------------------------------------------------------------- self-loop + bias init
// out[i,d] = xw[i,d] * dinv[i]^2 + b[d]   (fully overwrites d_out)
__global__ __launch_bounds__(256) void init_out_kernel(const float* __restrict__ xw,
                                                       const float* __restrict__ dinv,
                                                       const float* __restrict__ bias,
                                                       float* __restrict__ out,
                                                       int N) {
    const int total4 = N * (D_OUT / 4);
    for (int idx = blockIdx.x * blockDim.x + threadIdx.x; idx < total4;
         idx += gridDim.x * blockDim.x) {
        int rowi = idx >> 7;            // / 128
        int d4   = idx & 127;
        float s  = dinv[rowi];
        s = s * s;
        float4 v = ((const float4*)xw)[idx];
        float4 b = ((const float4*)bias)[d4];
        v.x = v.x * s + b.x;
        v.y = v.y * s + b.y;
        v.z = v.z * s + b.z;
        v.w = v.w * s + b.w;
        ((float4*)out)[idx] = v;
    }
}

// ---------------------------------------------------------------- edge scatter
// One wave per edge: float4 gathers from L2-resident xw, coalesced f32 atomics.
__global__ __launch_bounds__(256) void edge_scatter_kernel(const int* __restrict__ ei,
                                                           const float* __restrict__ ew,
                                                           const float* __restrict__ dinv,
                                                           const float* __restrict__ xw,
                                                           float* __restrict__ out,
                                                           int E) {
    int e = blockIdx.x * 8 + (threadIdx.x >> 5);
    if (e >= E) return;
    int r = ei[e];        // source (gather row)
    int c = ei[E + e];    // target (scatter row)
    float nrm = dinv[r] * ew[e] * dinv[c];

    int lane = threadIdx.x & 31;
    const float4* src = (const float4*)(xw + (size_t)r * D_OUT);
    float* dst = out + (size_t)c * D_OUT;

    #pragma unroll
    for (int i = 0; i < 4; ++i) {
        int q = lane + i * 32;          // float4 index 0..127
        float4 v = src[q];
        int d = q * 4;
        atomicAdd(dst + d + 0, v.x * nrm);
        atomicAdd(dst + d + 1, v.y * nrm);
        atomicAdd(dst + d + 2, v.z * nrm);
        atomicAdd(dst + d + 3, v.w * nrm);
    }
}

// ---------------------------------------------------------------- batchnorm stats
__global__ __launch_bounds__(512) void bn_stats_kernel(const float* __restrict__ out,
                                                       float* __restrict__ colsum,
                                                       float* __restrict__ colsumsq,
                                                       int N, int rows_per_block) {
    int col = threadIdx.x;  // blockDim = 512 = D_OUT
    int r0 = blockIdx.x * rows_per_block;
    int r1 = r0 + rows_per_block;
    if (r1 > N) r1 = N;
    float s = 0.0f, s2 = 0.0f;
    for (int r = r0; r < r1; ++r) {
        float v = out[(size_t)r * D_OUT + col];
        s  += v;
        s2 += v * v;
    }
    atomicAdd(&colsum[col], s);
    atomicAdd(&colsumsq[col], s2);
}

__global__ __launch_bounds__(512) void bn_finalize_kernel(const float* __restrict__ colsum,
                                                          const float* __restrict__ colsumsq,
                                                          const float* __restrict__ gamma,
                                                          const float* __restrict__ beta,
                                                          float* __restrict__ scale,
                                                          float* __restrict__ shift,
                                                          int N) {
    int t = threadIdx.x;  // blockDim = 512
    float invN = 1.0f / (float)N;
    float mean = colsum[t] * invN;
    float var  = colsumsq[t] * invN - mean * mean;
    if (var < 0.0f) var = 0.0f;
    float istd = rsqrtf(var + BN_EPS);
    float g    = gamma[t];
    scale[t] = g * istd;
    shift[t] = beta[t] - mean * g * istd;
}

__global__ __launch_bounds__(256) void bn_apply_kernel(float* __restrict__ out,
                                                       const float* __restrict__ scale,
                                                       const float* __restrict__ shift,
                                                       int N) {
    const int total4 = N * (D_OUT / 4);
    for (int idx = blockIdx.x * blockDim.x + threadIdx.x; idx < total4;
         idx += gridDim.x * blockDim.x) {
        int d4 = idx & 127;
        float4 v  = ((float4*)out)[idx];
        float4 sc = ((const float4*)scale)[d4];
        float4 sh = ((const float4*)shift)[d4];
        v.x = v.x * sc.x + sh.x;
        v.y = v.y * sc.y + sh.y;
        v.z = v.z * sc.z + sh.z;
        v.w = v.w * sc.w + sh.w;
        ((float4*)out)[idx] = v;
    }
}

// ---------------------------------------------------------------- launch
extern "C" void kernel_launch(void* const* d_in, const int* in_sizes, int n_in,
                              void* d_out, int out_size, void* d_ws, size_t ws_size,
                              hipStream_t stream) {
    const float* x     = (const float*)d_in[0];
    const int*   ei    = (const int*)d_in[1];
    const float* ew    = (const float*)d_in[2];
    const float* W     = (const float*)d_in[3];
    const float* bias  = (const float*)d_in[4];
    const float* gamma = (const float*)d_in[5];
    const float* beta  = (const float*)d_in[6];
    float* out = (float*)d_out;

    const int N = in_sizes[0] / D_IN;   // 50000
    const int E = in_sizes[2];          // 800000

    // workspace layout
    float* deg      = (float*)d_ws;                               // N (reused as dinv)
    size_t degPad   = ((size_t)N + 255) & ~(size_t)255;
    float* xw       = deg + degPad;                               // N * 512
    float* colsum   = xw + (size_t)N * D_OUT;                     // 512
    float* colsumsq = colsum + D_OUT;                             // 512
    float* scale    = colsumsq + D_OUT;                           // 512
    float* shift    = scale + D_OUT;                              // 512

    // 1) degree with self loops
    deg_init_kernel<<<(N + 255) / 256, 256, 0, stream>>>(deg, N);
    stats_init_kernel<<<1, 512, 0, stream>>>(colsum, colsumsq);
    deg_scatter_kernel<<<(E + 255) / 256, 256, 0, stream>>>(ei, ew, deg, E);
    deg_invsqrt_kernel<<<(N + 255) / 256, 256, 0, stream>>>(deg, N);

    // 2) xw = x @ W  (fp32 WMMA, 64x64 wave tiles)
    dim3 ggrid((N + 63) / 64, D_OUT / 64);
    gemm_wmma_kernel<<<ggrid, 32, 0, stream>>>(x, W, xw, N);

    // 3) out = self-loop term + bias, then edge scatter-add
    init_out_kernel<<<4096, 256, 0, stream>>>(xw, deg, bias, out, N);
    edge_scatter_kernel<<<(E + 7) / 8, 256, 0, stream>>>(ei, ew, deg, xw, out, E);

    // 4) batchnorm
    const int rows_per_block = 128;
    bn_stats_kernel<<<(N + rows_per_block - 1) / rows_per_block, 512, 0, stream>>>(
        out, colsum, colsumsq, N, rows_per_block);
    bn_finalize_kernel<<<1, 512, 0, stream>>>(colsum, colsumsq, gamma, beta, scale, shift, N);
    bn_apply_kernel<<<4096, 256, 0, stream>>>(out, scale, shift, N);
}